// GrassmannianLayerSelector_23235773071394
// MI455X (gfx1250) — compile-verified
//
#include <hip/hip_runtime.h>
#include <hip/hip_bf16.h>

// ---------------------------------------------------------------------------
// Problem constants (match reference)
// ---------------------------------------------------------------------------
constexpr int PN = 4;        // student layers
constexpr int LT = 12;       // teacher layers
constexpr int BN = 16;       // batch
constexpr int NN = 197;      // tokens
constexpr int DS = 384;      // student dim (= projected dim)
constexpr int DT = 768;      // teacher dim
constexpr int KR = 96;       // subspace rank = DS/4
constexpr long MROWS = (long)BN * NN;          // 3152 = 197*16
constexpr float EPS_REG = 1e-4f;
constexpr float CLAMP = 1.0f - 1e-7f;

typedef __attribute__((ext_vector_type(2))) float v2f;
typedef __attribute__((ext_vector_type(8))) float v8f;
typedef unsigned int u32x4 __attribute__((ext_vector_type(4)));
typedef int i32x4 __attribute__((ext_vector_type(4)));
typedef int i32x8 __attribute__((ext_vector_type(8)));

// ---------------------------------------------------------------------------
// TDM-staged WMMA GEMM:  C[MxN] = A[MxK] * B[NxK]^T   (both row-major)
//   - workgroup = 128 threads (4 waves); C tile = 16(M) x 64(N)
//   - Tensor Data Mover stages A(16xKT) and B(64xKT) f32 tiles into LDS,
//     with TDM padding (1 dword per 32) -> 33-dword LDS row stride,
//     conflict-free ds reads
//   - wave 0 issues tensor_load_to_lds, waits TENSORcnt, barrier releases all
// ---------------------------------------------------------------------------
constexpr int KT = 32;  // K-tile (dwords per row)

__global__ __launch_bounds__(128) void k_gemm_tdm(const float* __restrict__ A,
                                                  const float* __restrict__ Bm,
                                                  float* __restrict__ C,
                                                  int Kd, int Arows, int Brows,
                                                  int ldc) {
  __shared__ float As[16][KT + 1];   // TDM pads 1 dword per 32 stored
  __shared__ float Bs[64][KT + 1];
  const int tid = threadIdx.x;
  const int wave = tid >> 5;
  const int lane = tid & 31;
  const int half = lane >> 4;
  const int l16 = lane & 15;
  const int m0 = blockIdx.y * 16;
  const int n0 = blockIdx.x * 64;

  // D# group 1 (ISA 8.4): data_size=2 (4B), pad_enable, pad_interval=4 (32
  // dwords), pad_amount=0 (1 dword); tensor_dim0=Kd, tensor_dim1=rows,
  // tile_dim0=KT, tile_dim1=16/64, tensor_dim0_stride=Kd
  const unsigned g1w0 = (2u << 16) | (1u << 20) | (4u << 22);
  i32x8 g1A;
  g1A[0] = (int)g1w0;
  g1A[1] = (int)(((unsigned)Kd & 0xFFFFu) << 16);
  g1A[2] = (int)(((unsigned)Kd >> 16) | (((unsigned)Arows & 0xFFFFu) << 16));
  g1A[3] = (int)(((unsigned)Arows >> 16) | ((unsigned)KT << 16));
  g1A[4] = 16;          // tile_dim1 = 16 rows, tile_dim2 = 0
  g1A[5] = Kd;          // stride0 low 32
  g1A[6] = 0;
  g1A[7] = 0;
  i32x8 g1B = g1A;
  g1B[2] = (int)(((unsigned)Kd >> 16) | (((unsigned)Brows & 0xFFFFu) << 16));
  g1B[3] = (int)(((unsigned)Brows >> 16) | ((unsigned)KT << 16));
  g1B[4] = 64;          // tile_dim1 = 64 rows
  const i32x4 gz = {0, 0, 0, 0};
  const i32x8 gz8 = {0, 0, 0, 0, 0, 0, 0, 0};

  const unsigned ldsA = (unsigned)(uintptr_t)&As[0][0];
  const unsigned ldsB = (unsigned)(uintptr_t)&Bs[0][0];
  const unsigned long long baseA =
      (unsigned long long)(uintptr_t)A + (unsigned long long)m0 * Kd * 4ull;
  const unsigned long long baseB =
      (unsigned long long)(uintptr_t)Bm + (unsigned long long)n0 * Kd * 4ull;

  v8f acc = {0.f, 0.f, 0.f, 0.f, 0.f, 0.f, 0.f, 0.f};

  for (int k = 0; k < Kd; k += KT) {
    __syncthreads();  // previous tile consumed before overwrite
    if (wave == 0) {  // wave-uniform branch; TDM ignores EXEC
      const unsigned long long ga = baseA + (unsigned long long)k * 4ull;
      const unsigned long long gb = baseB + (unsigned long long)k * 4ull;
      // D# group 0: count=1, lds_addr, global_addr[56:0], type=2
      u32x4 g0A = {1u, ldsA, (unsigned)ga,
                   (unsigned)((ga >> 32) & 0x1FFFFFFull) | (2u << 30)};
      u32x4 g0B = {1u, ldsB, (unsigned)gb,
                   (unsigned)((gb >> 32) & 0x1FFFFFFull) | (2u << 30)};
      __builtin_amdgcn_tensor_load_to_lds(g0A, g1A, gz, gz, gz8, 0);
      __builtin_amdgcn_tensor_load_to_lds(g0B, g1B, gz, gz, gz8, 0);
      __builtin_amdgcn_s_wait_tensorcnt(0);
    }
    __syncthreads();  // LDS tiles visible to all 4 waves
#pragma unroll
    for (int kk = 0; kk < KT; kk += 4) {
      const int c = kk + half * 2;
      v2f a, b;
      a.x = As[l16][c];             a.y = As[l16][c + 1];
      b.x = Bs[wave * 16 + l16][c]; b.y = Bs[wave * 16 + l16][c + 1];
      acc = __builtin_amdgcn_wmma_f32_16x16x4_f32(false, a, false, b,
                                                  (short)0, acc, false, false);
    }
  }

#pragma unroll
  for (int r = 0; r < 8; ++r) {
    C[(long)(m0 + r + half * 8) * ldc + (n0 + wave * 16 + l16)] = acc[r];
  }
}

// ---------------------------------------------------------------------------
// Generic batched WMMA f32 GEMM (16x16 tile per wave, K-step 4)
// MODE 1: C[MxN] = A[MxK] * B[KxN]       (A K-contig, B N-contig)
// MODE 2: C[MxN] = A[KxM]^T * B[KxN]     (both N/M-contig)
// PAIR=true (with MODE 2): batch q -> A batch (12 + q/12), B batch (q%12)
// ---------------------------------------------------------------------------
enum { GEMM_AB = 1, GEMM_ATB = 2 };

template <int MODE, bool PAIR>
__global__ __launch_bounds__(32) void k_gemm(const float* __restrict__ A,
                                             const float* __restrict__ Bm,
                                             float* __restrict__ C,
                                             int Kdim, int lda, int ldb, int ldc,
                                             long bsA, long bsB, long bsC) {
  const int batch = blockIdx.z;
  long offA, offB;
  if (PAIR) {
    const int p = batch / 12, l = batch % 12;
    offA = (long)(12 + p) * bsA;
    offB = (long)l * bsB;
  } else {
    offA = (long)batch * bsA;
    offB = (long)batch * bsB;
  }
  const float* Ab = A + offA;
  const float* Bb = Bm + offB;
  float* Cb = C + (long)batch * bsC;

  const int lane = threadIdx.x;      // one full wave, EXEC all ones
  const int half = lane >> 4;
  const int l16 = lane & 15;
  const int m0 = blockIdx.y * 16, n0 = blockIdx.x * 16;

  v8f acc = {0.f, 0.f, 0.f, 0.f, 0.f, 0.f, 0.f, 0.f};

  for (int k = 0; k < Kdim; k += 4) {
    const int ka = k + half * 2;
    v2f a, b;
    if (MODE == GEMM_AB) {
      const float* pa = Ab + (long)(m0 + l16) * lda + ka;
      a.x = pa[0]; a.y = pa[1];
      const float* pb = Bb + (long)ka * ldb + (n0 + l16);
      b.x = pb[0]; b.y = pb[ldb];
    } else {  // GEMM_ATB
      const float* pa = Ab + (long)ka * lda + (m0 + l16);
      a.x = pa[0]; a.y = pa[lda];
      const float* pb = Bb + (long)ka * ldb + (n0 + l16);
      b.x = pb[0]; b.y = pb[ldb];
    }
    acc = __builtin_amdgcn_wmma_f32_16x16x4_f32(false, a, false, b,
                                                (short)0, acc, false, false);
  }

#pragma unroll
  for (int r = 0; r < 8; ++r) {
    Cb[(long)(m0 + r + half * 8) * ldc + (n0 + l16)] = acc[r];
  }
}

// ---------------------------------------------------------------------------
// Column means of z [Mrows x d]
// ---------------------------------------------------------------------------
__global__ void k_colmean(const float* __restrict__ z, float* __restrict__ mean,
                          int mrows, int d) {
  const int j = blockIdx.x * blockDim.x + threadIdx.x;
  if (j >= d) return;
  float s = 0.f;
  for (int m = 0; m < mrows; ++m) s += z[(long)m * d + j];
  mean[j] = s / (float)mrows;
}

// cov = G/M - mean*mean^T + eps*I
__global__ void k_cov_finalize(const float* __restrict__ G,
                               const float* __restrict__ mean,
                               float* __restrict__ cov, float invM) {
  const int idx = blockIdx.x * blockDim.x + threadIdx.x;
  if (idx >= DS * DS) return;
  const int i = idx / DS, j = idx % DS;
  float v = G[idx] * invM - mean[i] * mean[j];
  if (i == j) v += EPS_REG;
  cov[idx] = v;
}

// U0 = [I_96; 0] + tiny deterministic perturbation (16 batches)
__global__ void k_initU(float* __restrict__ U) {
  const long total = 16L * DS * KR;
  const long idx = (long)blockIdx.x * blockDim.x + threadIdx.x;
  if (idx >= total) return;
  const int j = (int)(idx % KR);
  const int i = (int)((idx / KR) % DS);
  unsigned h = (unsigned)idx * 2654435761u;
  h ^= h >> 16; h *= 2246822519u; h ^= h >> 13;
  const float noise = (float)(h & 1023u) * 1e-6f;
  U[idx] = ((i == j) ? 1.0f : 0.0f) + noise;
}

// ---------------------------------------------------------------------------
// Cholesky of 96x96 SPD matrix in LDS (one workgroup per batch)
// ---------------------------------------------------------------------------
__global__ __launch_bounds__(96) void k_chol(const float* __restrict__ G,
                                             float* __restrict__ L) {
  __shared__ float S[KR * KR];
  const int t = threadIdx.x;
  const int b = blockIdx.x;
  const float* Gb = G + (long)b * KR * KR;
  for (int i = t; i < KR * KR; i += blockDim.x) S[i] = Gb[i];
  __syncthreads();
  for (int k = 0; k < KR; ++k) {
    if (t == 0) S[k * KR + k] = sqrtf(fmaxf(S[k * KR + k], 1e-20f));
    __syncthreads();
    const float dk = S[k * KR + k];
    if (t > k) S[t * KR + k] /= dk;
    __syncthreads();
    if (t > k) {
      const float ltk = S[t * KR + k];
      for (int i = k + 1; i <= t; ++i) S[t * KR + i] -= ltk * S[i * KR + k];
    }
    __syncthreads();
  }
  float* Lb = L + (long)b * KR * KR;
  for (int i = t; i < KR * KR; i += blockDim.x) {
    const int r = i / KR, c = i % KR;
    Lb[i] = (c <= r) ? S[i] : 0.0f;
  }
}

// Solve U * L^T = W row-wise (forward substitution, L staged in LDS)
__global__ __launch_bounds__(64) void k_trisolve(const float* __restrict__ W,
                                                 const float* __restrict__ L,
                                                 float* __restrict__ U) {
  __shared__ float Ls[KR * KR];
  const int b = blockIdx.y;
  const int row = blockIdx.x * 64 + threadIdx.x;  // 0..383
  const float* Lb = L + (long)b * KR * KR;
  for (int i = threadIdx.x; i < KR * KR; i += 64) Ls[i] = Lb[i];
  __syncthreads();
  const float* w = W + (long)b * DS * KR + (long)row * KR;
  float y[KR];
  for (int k = 0; k < KR; ++k) {
    float acc = w[k];
    for (int j = 0; j < k; ++j) acc -= Ls[k * KR + j] * y[j];
    y[k] = acc / Ls[k * KR + k];
  }
  float* u = U + (long)b * DS * KR + (long)row * KR;
  for (int k = 0; k < KR; ++k) u[k] = y[k];
}

// ---------------------------------------------------------------------------
// Jacobi eigenvalues of symmetric 96x96 (= sigma^2 of M); emit sum(theta^2)
// ---------------------------------------------------------------------------
__global__ __launch_bounds__(96) void k_eig96(const float* __restrict__ T,
                                              float* __restrict__ dgr) {
  __shared__ float S[KR * KR];
  __shared__ float red[KR];
  const int t = threadIdx.x;
  const int b = blockIdx.x;
  const float* Tb = T + (long)b * KR * KR;
  for (int i = t; i < KR * KR; i += blockDim.x) S[i] = Tb[i];
  __syncthreads();
  for (int sweep = 0; sweep < 6; ++sweep) {
    for (int p = 0; p < KR - 1; ++p) {
      for (int q = p + 1; q < KR; ++q) {
        const float app = S[p * KR + p];
        const float aqq = S[q * KR + q];
        const float apq = S[p * KR + q];
        float c = 1.f, s = 0.f;
        if (fabsf(apq) > 1e-12f) {  // uniform branch (same LDS values)
          const float tau = (aqq - app) / (2.f * apq);
          const float tt = ((tau >= 0.f) ? 1.f : -1.f) /
                           (fabsf(tau) + sqrtf(1.f + tau * tau));
          c = 1.f / sqrtf(1.f + tt * tt);
          s = tt * c;
        }
        __syncthreads();
        const float sip = S[t * KR + p], siq = S[t * KR + q];
        if (t != p && t != q) {
          const float nip = c * sip - s * siq;
          const float niq = s * sip + c * siq;
          S[t * KR + p] = nip; S[p * KR + t] = nip;
          S[t * KR + q] = niq; S[q * KR + t] = niq;
        }
        __syncthreads();
        if (t == 0) {
          S[p * KR + p] = c * c * app - 2.f * s * c * apq + s * s * aqq;
          S[q * KR + q] = s * s * app + 2.f * s * c * apq + c * c * aqq;
          S[p * KR + q] = 0.f; S[q * KR + p] = 0.f;
        }
        __syncthreads();
      }
    }
  }
  const float lam = fmaxf(S[t * KR + t], 0.f);
  const float sig = fminf(sqrtf(lam), CLAMP);
  const float th = acosf(sig);
  red[t] = th * th;
  __syncthreads();
  if (t == 0) {
    float sum = 0.f;
    for (int i = 0; i < KR; ++i) sum += red[i];
    dgr[b] = sum;
  }
}

// weights[p][l] = softmax_l(-d[p][l] / (K * softplus(logt[p])))
__global__ void k_weights(const float* __restrict__ dgr,
                          const float* __restrict__ logt,
                          float* __restrict__ w) {
  const int p = threadIdx.x;
  if (p >= PN) return;
  const float x = logt[p];
  const float tau = (x > 20.f) ? x : log1pf(expf(x));
  const float inv = 1.f / ((float)KR * tau);
  float lg[LT];
  float mx = -1e30f;
  for (int l = 0; l < LT; ++l) { lg[l] = -dgr[p * LT + l] * inv; mx = fmaxf(mx, lg[l]); }
  float sum = 0.f;
  for (int l = 0; l < LT; ++l) { lg[l] = expf(lg[l] - mx); sum += lg[l]; }
  for (int l = 0; l < LT; ++l) w[p * LT + l] = lg[l] / sum;
}

// ---------------------------------------------------------------------------
// Bandwidth-bound mix: out[p] = sum_l w[p][l] * teacher[l]  (float4 streaming,
// every teacher element read once, all P outputs produced per element)
// ---------------------------------------------------------------------------
__global__ __launch_bounds__(256) void k_mix(const float* __restrict__ teach,
                                             const float* __restrict__ w,
                                             float* __restrict__ out,
                                             long perLayer4) {
  __shared__ float ws[PN * LT];
  if (threadIdx.x < PN * LT) ws[threadIdx.x] = w[threadIdx.x];
  __syncthreads();
  const long idx = (long)blockIdx.x * blockDim.x + threadIdx.x;
  if (idx >= perLayer4) return;
  const float4* tp = (const float4*)teach;
  float4 acc[PN];
#pragma unroll
  for (int p = 0; p < PN; ++p) acc[p] = make_float4(0.f, 0.f, 0.f, 0.f);
#pragma unroll
  for (int l = 0; l < LT; ++l) {
    const float4 v = tp[(long)l * perLayer4 + idx];
#pragma unroll
    for (int p = 0; p < PN; ++p) {
      const float wp = ws[p * LT + l];
      acc[p].x += wp * v.x; acc[p].y += wp * v.y;
      acc[p].z += wp * v.z; acc[p].w += wp * v.w;
    }
  }
  float4* op = (float4*)out;
#pragma unroll
  for (int p = 0; p < PN; ++p) op[(long)p * perLayer4 + idx] = acc[p];
}

// ---------------------------------------------------------------------------
// Host orchestration
// ---------------------------------------------------------------------------
extern "C" void kernel_launch(void* const* d_in, const int* in_sizes, int n_in,
                              void* d_out, int out_size, void* d_ws, size_t ws_size,
                              hipStream_t stream) {
  const float* student = (const float*)d_in[0];  // [4,16,197,384]
  const float* teacher = (const float*)d_in[1];  // [12,16,197,768]
  const float* attns   = (const float*)d_in[2];  // [12,16,12,197,197]
  const float* proj_s  = (const float*)d_in[3];  // [384,384]
  const float* proj_t  = (const float*)d_in[4];  // [384,768]
  const float* logt    = (const float*)d_in[5];  // [4]
  float* out = (float*)d_out;

  // --- workspace layout (floats) ---
  float* ws = (float*)d_ws;
  constexpr long SZ_Z    = MROWS * DS;          // reused per layer
  constexpr long SZ_MEAN = 16L * DS;
  constexpr long SZ_COV  = 16L * DS * DS;
  constexpr long SZ_U    = 16L * DS * KR;
  constexpr long SZ_G96  = 16L * KR * KR;
  constexpr long SZ_PAIR = 48L * KR * KR;
  float* zbuf  = ws;                 long off = SZ_Z;
  float* meanb = ws + off;           off += SZ_MEAN;
  float* covA  = ws + off;           off += SZ_COV;
  float* covB  = ws + off;           off += SZ_COV;
  float* Ubuf  = ws + off;           off += SZ_U;
  float* Wbuf  = ws + off;           off += SZ_U;
  float* G96   = ws + off;           off += SZ_G96;
  float* Lch   = ws + off;           off += SZ_G96;
  float* Mpair = ws + off;           off += SZ_PAIR;
  float* Tpair = ws + off;           off += SZ_PAIR;
  float* dgr   = ws + off;           off += 48;
  float* wts   = ws + off;           off += 48;

  constexpr long COVS = (long)DS * DS;
  constexpr long US   = (long)DS * KR;
  constexpr long KKS  = (long)KR * KR;
  constexpr long TOK_L = MROWS * DT;
  constexpr long STU_L = MROWS * DS;

  const dim3 wave(32);

  // 1) Per "layer" b (0-11 teacher, 12-15 student): z, mean, Gram, cov
  for (int b = 0; b < 16; ++b) {
    const bool isT = (b < LT);
    const float* X = isT ? (teacher + (long)b * TOK_L)
                         : (student + (long)(b - LT) * STU_L);
    const float* Pm = isT ? proj_t : proj_s;
    const int Kd = isT ? DT : DS;
    // z[MROWS x DS] = X[MROWS x Kd] * P[DS x Kd]^T  (TDM + WMMA)
    k_gemm_tdm<<<dim3(DS / 64, (int)(MROWS / 16), 1), dim3(128), 0, stream>>>(
        X, Pm, zbuf, Kd, (int)MROWS, DS, DS);
    // column means
    k_colmean<<<dim3((DS + 127) / 128), dim3(128), 0, stream>>>(
        zbuf, meanb + (long)b * DS, (int)MROWS, DS);
    // G = z^T z  (WMMA, ATB with A==B)
    k_gemm<GEMM_ATB, false><<<dim3(DS / 16, DS / 16, 1), wave, 0, stream>>>(
        zbuf, zbuf, covB + (long)b * COVS, (int)MROWS, DS, DS, DS, 0, 0, 0);
    // cov = G/M - mean mean^T + eps I
    k_cov_finalize<<<dim3((DS * DS + 255) / 256), dim3(256), 0, stream>>>(
        covB + (long)b * COVS, meanb + (long)b * DS, covA + (long)b * COVS,
        1.0f / (float)MROWS);
  }

  // 2) Spectral-gap amplification: covB = cov^8 (three squarings, WMMA)
  k_gemm<GEMM_AB, false><<<dim3(DS / 16, DS / 16, 16), wave, 0, stream>>>(
      covA, covA, covB, DS, DS, DS, DS, COVS, COVS, COVS);       // cov^2
  k_gemm<GEMM_AB, false><<<dim3(DS / 16, DS / 16, 16), wave, 0, stream>>>(
      covB, covB, covA, DS, DS, DS, DS, COVS, COVS, COVS);       // cov^4
  k_gemm<GEMM_AB, false><<<dim3(DS / 16, DS / 16, 16), wave, 0, stream>>>(
      covA, covA, covB, DS, DS, DS, DS, COVS, COVS, COVS);       // cov^8

  // 3) Subspace iteration with Cholesky-QR (top-96 invariant subspace;
  //    principal angles are basis-invariant, so Chol-QR basis is exact)
  k_initU<<<dim3((int)((16L * DS * KR + 255) / 256)), dim3(256), 0, stream>>>(Ubuf);
  for (int it = 0; it < 12; ++it) {
    // W = cov^8 * U  (WMMA)
    k_gemm<GEMM_AB, false><<<dim3(KR / 16, DS / 16, 16), wave, 0, stream>>>(
        covB, Ubuf, Wbuf, DS, DS, KR, KR, COVS, US, US);
    // G = W^T W  (WMMA)
    k_gemm<GEMM_ATB, false><<<dim3(KR / 16, KR / 16, 16), wave, 0, stream>>>(
        Wbuf, Wbuf, G96, DS, KR, KR, KR, US, US, KKS);
    // L = chol(G); U = W * L^-T
    k_chol<<<dim3(16), dim3(96), 0, stream>>>(G96, Lch);
    k_trisolve<<<dim3(DS / 64, 16), dim3(64), 0, stream>>>(Wbuf, Lch, Ubuf);
  }

  // 4) M[p,l] = U_s^T U_t  (WMMA, pair-indexed batches)
  k_gemm<GEMM_ATB, true><<<dim3(KR / 16, KR / 16, 48), wave, 0, stream>>>(
      Ubuf, Ubuf, Mpair, DS, KR, KR, KR, US, US, KKS);
  // 5) T = M^T M  (WMMA); sigma^2 = eig(T)
  k_gemm<GEMM_ATB, false><<<dim3(KR / 16, KR / 16, 48), wave, 0, stream>>>(
      Mpair, Mpair, Tpair, KR, KR, KR, KR, KKS, KKS, KKS);
  k_eig96<<<dim3(48), dim3(96), 0, stream>>>(Tpair, dgr);

  // 6) softmax weights
  k_weights<<<dim3(1), dim3(32), 0, stream>>>(dgr, logt, wts);

  // 7) streaming mixes (bandwidth-bound: ~0.48 GB total traffic -> ~20 us)
  constexpr long TOK4 = TOK_L / 4;                        // 605,184
  constexpr long ATT_L = (long)BN * 12 * NN * NN;         // 7,451,328
  constexpr long ATT4 = ATT_L / 4;                        // 1,862,832
  k_mix<<<dim3((int)((TOK4 + 255) / 256)), dim3(256), 0, stream>>>(
      teacher, wts, out, TOK4);
  k_mix<<<dim3((int)((ATT4 + 255) / 256)), dim3(256), 0, stream>>>(
      attns, wts, out + (long)PN * TOK_L, ATT4);
}